// MBSC_6201932775871
// MI455X (gfx1250) — compile-verified
//
#include <hip/hip_runtime.h>
#include <hip/hip_bf16.h>
#include <math.h>

// CDNA5 (gfx1250) wave32 WMMA types
typedef __attribute__((ext_vector_type(2))) float v2f;   // A/B frag for 16x16x4 f32 (2 VGPRs)
typedef __attribute__((ext_vector_type(8))) float v8f;   // C/D frag (8 VGPRs)

#define NPTS        32768
#define K_CLUSTERS  64
#define M_PTS       512                 // points per cluster
#define NB          32                  // 16x16 tile blocks per side (512/16)
#define D_THRE      0.03f
#define INV_D2      (1.0f / (D_THRE * D_THRE))

// One workgroup per cluster. 256 threads = 8 wave32.
// Gram-matrix formulation: dist^2(i,j) = ns[i] + ns[j] - 2*dot(i,j),
// dot() computed 16x16-tile-at-a-time with v_wmma_f32_16x16x4_f32.
// Symmetry: process only bj >= bi, weight off-diagonal tiles by 2.
__global__ __launch_bounds__(256) void sc_cluster_kernel(
    const float* __restrict__ pc1, const float* __restrict__ flow,
    float* __restrict__ cluster_sums)
{
    __shared__ float S[M_PTS * 4];   // src points, padded K=4 (4th col = 0)
    __shared__ float T[M_PTS * 4];   // deformed points
    __shared__ float ns[M_PTS];      // |s_i|^2
    __shared__ float nt[M_PTS];      // |t_i|^2
    __shared__ float red[256];

    const int c   = blockIdx.x;
    const int tid = threadIdx.x;

    // Stage cluster into LDS: deform, pad, squared norms.
    for (int p = tid; p < M_PTS; p += 256) {
        const int g = (c * M_PTS + p) * 3;
        const float sx = pc1[g + 0], sy = pc1[g + 1], sz = pc1[g + 2];
        const float tx = sx + flow[g + 0], ty = sy + flow[g + 1], tz = sz + flow[g + 2];
        S[p * 4 + 0] = sx; S[p * 4 + 1] = sy; S[p * 4 + 2] = sz; S[p * 4 + 3] = 0.0f;
        T[p * 4 + 0] = tx; T[p * 4 + 1] = ty; T[p * 4 + 2] = tz; T[p * 4 + 3] = 0.0f;
        ns[p] = sx * sx + sy * sy + sz * sz;
        nt[p] = tx * tx + ty * ty + tz * tz;
    }
    __syncthreads();

    // Scalarize the wave id so tile ownership is a uniform (SCC) branch:
    // EXEC stays all-1s across every WMMA (hardware requirement).
    const int wv   = __builtin_amdgcn_readfirstlane(tid >> 5);
    const int lane = tid & 31;
    const int l16  = lane & 15;
    const int half = lane >> 4;          // 0 -> K={0,1}, 1 -> K={2,3} (ISA 16x4 f32 A layout)

    float acc = 0.0f;
    int   cnt = 0;                       // 528 upper-triangle tiles, 66 per wave

    for (int bi = 0; bi < NB; ++bi) {
        for (int bj = bi; bj < NB; ++bj, ++cnt) {
            if ((cnt & 7) != wv) continue;

            const int row = bi * 16 + l16;   // A: M = lane%16
            const int col = bj * 16 + l16;   // B = S^T: load row `col` of S

            v2f aS, bS, aT, bT;
            aS.x = S[row * 4 + 2 * half]; aS.y = S[row * 4 + 2 * half + 1];
            bS.x = S[col * 4 + 2 * half]; bS.y = S[col * 4 + 2 * half + 1];
            aT.x = T[row * 4 + 2 * half]; aT.y = T[row * 4 + 2 * half + 1];
            bT.x = T[col * 4 + 2 * half]; bT.y = T[col * 4 + 2 * half + 1];

            v8f cS = {};
            v8f cT = {};
            cS = __builtin_amdgcn_wmma_f32_16x16x4_f32(false, aS, false, bS,
                                                       (short)0, cS, false, false);
            cT = __builtin_amdgcn_wmma_f32_16x16x4_f32(false, aT, false, bT,
                                                       (short)0, cT, false, false);

            // C/D layout: VGPR r, lane l -> M = r + 8*(l>=16), N = l%16
            const float nsj = ns[col];
            const float ntj = nt[col];
            const float w   = (bi == bj) ? 1.0f : 2.0f;
            float tacc = 0.0f;
#pragma unroll
            for (int r = 0; r < 8; ++r) {
                const int i = bi * 16 + r + 8 * half;
                const float sq_s = ns[i] + nsj - 2.0f * cS[r];
                const float sq_t = nt[i] + ntj - 2.0f * cT[r];
                // safe-sqrt: sqrt(max(sq,0)); raw v_sqrt_f32 (~1ulp) — error
                // ~1e-5 m on <=100 m distances vs 0.03 m threshold.
                const float ds = __builtin_amdgcn_sqrtf(fmaxf(sq_s, 0.0f));
                const float dt = __builtin_amdgcn_sqrtf(fmaxf(sq_t, 0.0f));
                const float d  = ds - dt;
                tacc += fminf(d * d * INV_D2, 1.0f);   // 1 - max(0, 1-x) == min(x,1)
            }
            acc += w * tacc;
        }
    }

    // Block reduction -> per-cluster sum
    red[tid] = acc;
    __syncthreads();
    for (int s = 128; s > 0; s >>= 1) {
        if (tid < s) red[tid] += red[tid + s];
        __syncthreads();
    }
    if (tid == 0) cluster_sums[c] = red[0];
}

// Final: sum 64 cluster sums, scale by 1/(M*M*K). Separate kernel keeps
// d_out writes deterministic (no atomics on a poisoned buffer).
__global__ void sc_reduce_kernel(const float* __restrict__ cluster_sums,
                                 float* __restrict__ out)
{
    __shared__ float sh[K_CLUSTERS];
    const int tid = threadIdx.x;
    sh[tid] = cluster_sums[tid];
    __syncthreads();
    for (int s = K_CLUSTERS / 2; s > 0; s >>= 1) {
        if (tid < s) sh[tid] += sh[tid + s];
        __syncthreads();
    }
    if (tid == 0)
        out[0] = sh[0] * (1.0f / ((float)M_PTS * (float)M_PTS * (float)K_CLUSTERS));
}

extern "C" void kernel_launch(void* const* d_in, const int* in_sizes, int n_in,
                              void* d_out, int out_size, void* d_ws, size_t ws_size,
                              hipStream_t stream)
{
    const float* flow = (const float*)d_in[0];   // [1, 32768, 3] f32
    const float* pc1  = (const float*)d_in[1];   // [1, 32768, 3] f32
    // d_in[2] = labels (contiguous blocks -> reshape; unused)
    // d_in[3] = num_clusters scalar (device side; K fixed to 64 per setup)
    float* ws  = (float*)d_ws;                    // 64 per-cluster sums
    float* out = (float*)d_out;                   // scalar f32 loss

    sc_cluster_kernel<<<K_CLUSTERS, 256, 0, stream>>>(pc1, flow, ws);
    sc_reduce_kernel<<<1, K_CLUSTERS, 0, stream>>>(ws, out);
}